// ELMo_28965259444762
// MI455X (gfx1250) — compile-verified
//
#include <hip/hip_runtime.h>
#include <math.h>

// ---------------- types / WMMA wrapper ----------------
typedef __attribute__((ext_vector_type(2))) float v2f;
typedef __attribute__((ext_vector_type(8))) float v8f;

__device__ __forceinline__ v8f wmma_f32x4(v2f a, v2f b, v8f c) {
  // D = A(16x4) * B(4x16) + C(16x16), fp32 exact
  return __builtin_amdgcn_wmma_f32_16x16x4_f32(false, a, false, b, (short)0, c,
                                               false, false);
}

__device__ __forceinline__ float sigmoidf_(float x) {
  return 1.0f / (1.0f + __expf(-x));
}

#define BB 32   // batch
#define TT 64   // time
#define HH 512  // hidden

// perm(r): time-major row (t*B+b) <-> batch-major row (b*T+t).
__device__ __forceinline__ int permrow(int r) { return (r & 31) * 64 + (r >> 5); }

// ---------------- register-blocked WMMA GEMM ----------------
// C[M,N] = A[M,K] @ W[N,K]^T + b1[N] + b2[N]
// Each wave: 32x64 macro-tile = 2 M-subtiles x NJ N-subtiles (NJ templated).
// OOB N columns: W row pointer clamped; garbage acc columns discarded at store.
template <int NJ>
__device__ __forceinline__ void gemm_macro_tile(
    const float* __restrict__ Arow0, const float* __restrict__ Arow1,
    const float* const* Wr, const int* ncs,
    const float* __restrict__ b1, const float* __restrict__ b2,
    float* __restrict__ C, int N, int K, int m_base, int lane, int permC) {
  v8f acc[2][NJ];
#pragma unroll
  for (int i = 0; i < 2; ++i)
#pragma unroll
    for (int j = 0; j < NJ; ++j) acc[i][j] = (v8f){0.f,0.f,0.f,0.f,0.f,0.f,0.f,0.f};

#pragma unroll 2
  for (int k0 = 0; k0 < K; k0 += 4) {
    v2f a0 = *(const v2f*)(Arow0 + k0);
    v2f a1 = *(const v2f*)(Arow1 + k0);
#pragma unroll
    for (int j = 0; j < NJ; ++j) {
      v2f w = *(const v2f*)(Wr[j] + k0);
      acc[0][j] = wmma_f32x4(a0, w, acc[0][j]);
      acc[1][j] = wmma_f32x4(a1, w, acc[1][j]);
    }
  }

  const int vrow = (lane >> 4) << 3;
#pragma unroll
  for (int j = 0; j < NJ; ++j) {
    int nc = ncs[j];
    if (nc < N) {
      float bias = (b1 ? b1[nc] : 0.f) + (b2 ? b2[nc] : 0.f);
#pragma unroll
      for (int i = 0; i < 2; ++i)
#pragma unroll
        for (int v = 0; v < 8; ++v) {
          int mr = m_base + i * 16 + v + vrow;
          int cr = permC ? permrow(mr) : mr;
          C[(size_t)cr * N + nc] = acc[i][j][v] + bias;
        }
    }
  }
}

__global__ __launch_bounds__(256) void wmma_gemm_kernel(
    const float* __restrict__ A, const float* __restrict__ W,
    const float* __restrict__ b1, const float* __restrict__ b2,
    float* __restrict__ C, int M, int N, int K, int permA, int permC) {
  const int lane = threadIdx.x;            // 0..31
  const int wave = threadIdx.y;            // 0..7
  const int tilesN = (N + 63) >> 6;
  const int tilesM = M >> 5;               // M always multiple of 32 here
  const int tile = blockIdx.x * 8 + wave;
  if (tile >= tilesM * tilesN) return;     // uniform per wave

  const int mt = tile / tilesN, nt = tile - mt * tilesN;
  const int m_base = mt << 5, n_base = nt << 6;
  const int l16 = lane & 15;
  const int kb = (lane >> 4) << 1;

  int ar0 = m_base + l16, ar1 = m_base + 16 + l16;
  if (permA) { ar0 = permrow(ar0); ar1 = permrow(ar1); }
  const float* Arow0 = A + (size_t)ar0 * K + kb;
  const float* Arow1 = A + (size_t)ar1 * K + kb;

  const float* Wr[4];
  int ncs[4];
#pragma unroll
  for (int j = 0; j < 4; ++j) {
    int nc = n_base + j * 16 + l16;
    ncs[j] = nc;
    int ncc = nc < N ? nc : N - 1;         // clamp address only; no masking needed
    Wr[j] = W + (size_t)ncc * K + kb;
  }

  int nj = (N - n_base + 15) >> 4;         // active 16-col subtiles (wave-uniform)
  if (nj >= 4)
    gemm_macro_tile<4>(Arow0, Arow1, Wr, ncs, b1, b2, C, N, K, m_base, lane, permC);
  else if (nj == 3)
    gemm_macro_tile<3>(Arow0, Arow1, Wr, ncs, b1, b2, C, N, K, m_base, lane, permC);
  else if (nj == 2)
    gemm_macro_tile<2>(Arow0, Arow1, Wr, ncs, b1, b2, C, N, K, m_base, lane, permC);
  else
    gemm_macro_tile<1>(Arow0, Arow1, Wr, ncs, b1, b2, C, N, K, m_base, lane, permC);
}

// ---------------- LSTM recurrent scan ----------------
// grid.x = 2 (branch f / b), 1024 threads = 32 waves.
// xg: [T, B, 4H] precomputed x@Wih^T+bih+bhh (time-major). Whh: [4H, H].
// y (hs): [T, B, H] time-major. Both batch halves share each W fragment.
__global__ __launch_bounds__(1024) void lstm_scan_kernel(
    const float* __restrict__ xg_f, const float* __restrict__ Whh_f,
    float* __restrict__ y_f, const float* __restrict__ xg_b,
    const float* __restrict__ Whh_b, float* __restrict__ y_b) {
  __shared__ float hbuf[2][BB * HH];  // 2 x 64KB double-buffered h_t

  const float* xg = (blockIdx.x == 0) ? xg_f : xg_b;
  const float* Whh = (blockIdx.x == 0) ? Whh_f : Whh_b;
  float* y = (blockIdx.x == 0) ? y_f : y_b;

  const int tid = threadIdx.x;
  const int lane = tid & 31;
  const int wave = tid >> 5;              // 0..31, owns h columns [16w, 16w+16)
  const int l16 = lane & 15;
  const int kb = (lane >> 4) << 1;
  const int col = (wave << 4) + l16;
  const int vrow = (lane >> 4) << 3;      // D-layout row offset

  for (int i = tid; i < BB * HH; i += 1024) hbuf[0][i] = 0.f;
  __syncthreads();

  const float* Wg0 = Whh + (size_t)(0 * HH + col) * HH + kb;
  const float* Wg1 = Whh + (size_t)(1 * HH + col) * HH + kb;
  const float* Wg2 = Whh + (size_t)(2 * HH + col) * HH + kb;
  const float* Wg3 = Whh + (size_t)(3 * HH + col) * HH + kb;

  v8f c0 = {0.f,0.f,0.f,0.f,0.f,0.f,0.f,0.f};
  v8f c1 = c0;

  for (int t = 0; t < TT; ++t) {
    const int cur = t & 1, nxt = cur ^ 1;
    const size_t xrow = (size_t)t * BB * (4 * HH);

    // prefetch next step's input gates (256KB = 1024 threads x 256B lines)
    if (t + 1 < TT)
      __builtin_prefetch(xg + xrow + (size_t)BB * 4 * HH + (size_t)tid * 64, 0, 1);

    // init 8 accumulators (4 gates x 2 batch halves) from precomputed xg
    v8f ai0, af0, ag0, ao0, ai1, af1, ag1, ao1;
#pragma unroll
    for (int v = 0; v < 8; ++v) {
      int br0 = v + vrow, br1 = 16 + v + vrow;
      size_t b0 = xrow + (size_t)br0 * (4 * HH);
      size_t b1x = xrow + (size_t)br1 * (4 * HH);
      ai0[v] = xg[b0 + 0 * HH + col];  ai1[v] = xg[b1x + 0 * HH + col];
      af0[v] = xg[b0 + 1 * HH + col];  af1[v] = xg[b1x + 1 * HH + col];
      ag0[v] = xg[b0 + 2 * HH + col];  ag1[v] = xg[b1x + 2 * HH + col];
      ao0[v] = xg[b0 + 3 * HH + col];  ao1[v] = xg[b1x + 3 * HH + col];
    }

#pragma unroll 2
    for (int k0 = 0; k0 < HH; k0 += 4) {
      v2f a0 = *(const v2f*)&hbuf[cur][l16 * HH + k0 + kb];        // ds_load_b64
      v2f a1 = *(const v2f*)&hbuf[cur][(16 + l16) * HH + k0 + kb];
      v2f w0 = *(const v2f*)(Wg0 + k0);
      v2f w1 = *(const v2f*)(Wg1 + k0);
      v2f w2 = *(const v2f*)(Wg2 + k0);
      v2f w3 = *(const v2f*)(Wg3 + k0);
      ai0 = wmma_f32x4(a0, w0, ai0);  ai1 = wmma_f32x4(a1, w0, ai1);
      af0 = wmma_f32x4(a0, w1, af0);  af1 = wmma_f32x4(a1, w1, af1);
      ag0 = wmma_f32x4(a0, w2, ag0);  ag1 = wmma_f32x4(a1, w2, ag1);
      ao0 = wmma_f32x4(a0, w3, ao0);  ao1 = wmma_f32x4(a1, w3, ao1);
    }

    // element-wise LSTM cell, both halves (PyTorch gate order i,f,g,o)
#pragma unroll
    for (int v = 0; v < 8; ++v) {
      {
        float iv = sigmoidf_(ai0[v]), fv = sigmoidf_(af0[v]);
        float gv = tanhf(ag0[v]),     ov = sigmoidf_(ao0[v]);
        float cv = fv * c0[v] + iv * gv; c0[v] = cv;
        float hv = ov * tanhf(cv);
        int br = v + vrow;
        y[(size_t)(t * BB + br) * HH + col] = hv;
        hbuf[nxt][br * HH + col] = hv;
      }
      {
        float iv = sigmoidf_(ai1[v]), fv = sigmoidf_(af1[v]);
        float gv = tanhf(ag1[v]),     ov = sigmoidf_(ao1[v]);
        float cv = fv * c1[v] + iv * gv; c1[v] = cv;
        float hv = ov * tanhf(cv);
        int br = 16 + v + vrow;
        y[(size_t)(t * BB + br) * HH + col] = hv;
        hbuf[nxt][br * HH + col] = hv;
      }
    }
    __syncthreads();  // h_{t+1} fully visible before next step reads it
  }
}

// ---------------- adaptive log-softmax (head + target cluster) ----------------
__global__ __launch_bounds__(256) void asoft_kernel(
    const float* __restrict__ hfin, const float* __restrict__ tp0,
    const float* __restrict__ tp1, const float* __restrict__ tp2,
    const float* __restrict__ head_w, const float* __restrict__ out0,
    const float* __restrict__ out1, const float* __restrict__ out2,
    const int* __restrict__ lab, float* __restrict__ out) {
  __shared__ float hrow[HH];
  __shared__ float hl[16];
  __shared__ float headlp[16];
  __shared__ float tprow[128];
  __shared__ float redm[256], reds[256];
  __shared__ float s_logit_t;

  const int s = blockIdx.x;
  const int tid = threadIdx.x;
  const int tgt = lab[s];

  for (int i = tid; i < HH; i += 256) hrow[i] = hfin[(size_t)s * HH + i];
  __syncthreads();

  if (tid < 13 * 16) {
    int j = tid >> 4, l = tid & 15;
    const float* w = head_w + j * HH;
    float p = 0.f;
    for (int k = l; k < HH; k += 16) p += hrow[k] * w[k];
    for (int off = 8; off; off >>= 1) p += __shfl_xor(p, off, 16);
    if (l == 0) hl[j] = p;
  }
  __syncthreads();
  if (tid == 0) {
    float mx = hl[0];
    for (int j = 1; j < 13; ++j) mx = fmaxf(mx, hl[j]);
    float sum = 0.f;
    for (int j = 0; j < 13; ++j) sum += __expf(hl[j] - mx);
    float lse = mx + __logf(sum);
    for (int j = 0; j < 13; ++j) headlp[j] = hl[j] - lse;
  }
  __syncthreads();

  if (tgt < 10) {  // shortlist word (tgt uniform across block)
    if (tid == 0) out[s] = headlp[tgt];
    return;
  }

  int ci, lo, osz, Kd;
  const float* tp; const float* ow;
  if (tgt < 100)       { ci = 0; lo = 10;   osz = 90;    Kd = 128; tp = tp0; ow = out0; }
  else if (tgt < 1000) { ci = 1; lo = 100;  osz = 900;   Kd = 32;  tp = tp1; ow = out1; }
  else                 { ci = 2; lo = 1000; osz = 49000; Kd = 8;   tp = tp2; ow = out2; }

  for (int i = tid; i < Kd; i += 256) tprow[i] = tp[(size_t)s * Kd + i];
  __syncthreads();

  float m = -INFINITY, ssum = 0.f;
  for (int j = tid; j < osz; j += 256) {
    const float* wr = ow + (size_t)j * Kd;
    float lg = 0.f;
    for (int k = 0; k < Kd; ++k) lg += tprow[k] * wr[k];
    if (j == tgt - lo) s_logit_t = lg;
    float nm = fmaxf(m, lg);
    ssum = ssum * __expf(m - nm) + __expf(lg - nm);
    m = nm;
  }
  redm[tid] = m; reds[tid] = ssum;
  __syncthreads();
  if (tid == 0) {
    float M = -INFINITY, S = 0.f;
    for (int i = 0; i < 256; ++i) {
      float nm = fmaxf(M, redm[i]);
      S = S * __expf(M - nm) + reds[i] * __expf(redm[i] - nm);
      M = nm;
    }
    out[s] = headlp[10 + ci] + s_logit_t - (M + __logf(S));
  }
}

// ---------------- host orchestration ----------------
extern "C" void kernel_launch(void* const* d_in, const int* in_sizes, int n_in,
                              void* d_out, int out_size, void* d_ws, size_t ws_size,
                              hipStream_t stream) {
  const float* vec  = (const float*)d_in[0];   // [B,T,E]
  const float* vec2 = (const float*)d_in[1];
  const int*   lab  = (const int*)d_in[2];     // [B*T]
  const int*   lab2 = (const int*)d_in[3];

  const float* P[40];
  int np = n_in - 4; if (np > 40) np = 40;
  for (int i = 0; i < np; ++i) P[i] = (const float*)d_in[4 + i];
  const float* head_w = P[28];
  const float* t0p = P[29]; const float* t0o = P[30];
  const float* t1p = P[31]; const float* t1o = P[32];
  const float* t2p = P[33]; const float* t2o = P[34];

  const int M = BB * TT;  // 2048
  float* ws = (float*)d_ws;
  size_t off = 0;
  auto alloc = [&](size_t n) { float* p = ws + off; off += n; return p; };
  float *x[2], *xg[2], *hs[2], *hfin[2], *tpb[2][3];
  for (int br = 0; br < 2; ++br) {
    x[br]      = alloc((size_t)M * HH);
    xg[br]     = alloc((size_t)M * 4 * HH);
    hs[br]     = alloc((size_t)M * HH);
    hfin[br]   = alloc((size_t)M * HH);
    tpb[br][0] = alloc((size_t)M * 128);
    tpb[br][1] = alloc((size_t)M * 32);
    tpb[br][2] = alloc((size_t)M * 8);
  }

  auto gemm = [&](const float* A, const float* W, const float* b1, const float* b2,
                  float* C, int m, int n, int k, int pA, int pC) {
    int tiles = (m / 32) * ((n + 63) / 64);
    dim3 blk(32, 8);
    dim3 grd((tiles + 7) / 8);
    wmma_gemm_kernel<<<grd, blk, 0, stream>>>(A, W, b1, b2, C, m, n, k, pA, pC);
  };

  const float* vv[2] = {vec, vec2};
  // embedding projection: x[t,b,:] = vec[b,t,:] @ emb_w^T + emb_b  (permA)
  for (int br = 0; br < 2; ++br)
    gemm(vv[br], P[br * 14 + 0], P[br * 14 + 1], nullptr, x[br], M, HH, 300, 1, 0);

  for (int li = 0; li < 2; ++li) {
    for (int br = 0; br < 2; ++br) {
      const float* const* Pb = P + br * 14 + 2 + li * 6;
      gemm(x[br], Pb[0], Pb[2], Pb[3], xg[br], M, 4 * HH, HH, 0, 0);
    }
    lstm_scan_kernel<<<dim3(2), dim3(1024), 0, stream>>>(
        xg[0], P[0 * 14 + 3 + li * 6], hs[0],
        xg[1], P[1 * 14 + 3 + li * 6], hs[1]);
    for (int br = 0; br < 2; ++br) {
      const float* const* Pb = P + br * 14 + 2 + li * 6;
      int last = (li == 1);
      gemm(hs[br], Pb[4], Pb[5], nullptr, last ? hfin[br] : x[br], M, HH, HH, 0, last);
    }
  }

  for (int br = 0; br < 2; ++br) {
    gemm(hfin[br], t0p, nullptr, nullptr, tpb[br][0], M, 128, HH, 0, 0);
    gemm(hfin[br], t1p, nullptr, nullptr, tpb[br][1], M, 32,  HH, 0, 0);
    gemm(hfin[br], t2p, nullptr, nullptr, tpb[br][2], M, 8,   HH, 0, 0);
  }

  const int* ll[2] = {lab, lab2};
  float* outp = (float*)d_out;
  for (int br = 0; br < 2; ++br) {
    asoft_kernel<<<dim3(M), dim3(256), 0, stream>>>(
        hfin[br], tpb[br][0], tpb[br][1], tpb[br][2], head_w, t0o, t1o, t2o,
        ll[br], outp + (size_t)br * M);
  }
}